// SAP_60756607369872
// MI455X (gfx1250) — compile-verified
//
#include <hip/hip_runtime.h>
#include <hip/hip_bf16.h>
#include <stdint.h>

// Problem constants (from the reference)
#define B_DIM   64
#define C_DIM   128
#define T_DIM   4096
#define N_ROW   (C_DIM * T_DIM)          // 524288 elements per row
#define K_SEL   52428                    // int(0.1 * N_ROW)

#define NBINS     4096
#define BIN_SHIFT 20                     // bin = ordered_u32 >> 20

#define THREADS          256             // 8 waves of 32 on CDNA5
#define VEC              4
#define ELEMS_PER_ITER   (THREADS * VEC) // 1024 floats / iter / block

#define CHUNKS1 16
#define CHUNK1  (N_ROW / CHUNKS1)            // 32768
#define ITERS1  (CHUNK1 / ELEMS_PER_ITER)    // 32

#define CHUNKS3 32
#define CHUNK3  (N_ROW / CHUNKS3)            // 16384
#define ITERS3  (CHUNK3 / ELEMS_PER_ITER)    // 16

typedef float v4f __attribute__((ext_vector_type(4)));
typedef int   v4i __attribute__((vector_size(16)));   // matches builtin param

// ---- CDNA5 async global -> LDS path (ASYNCcnt-tracked) ------------------
#if defined(__has_builtin)
#if __has_builtin(__builtin_amdgcn_global_load_async_to_lds_b128)
#define HAVE_ASYNC_B128 1
#endif
#if __has_builtin(__builtin_amdgcn_s_wait_asynccnt)
#define HAVE_WAIT_ASYNC 1
#endif
#endif

#if defined(HAVE_WAIT_ASYNC)
#define WAIT_ASYNC(n) __builtin_amdgcn_s_wait_asynccnt(n)
#elif defined(HAVE_ASYNC_B128)
#define WAIT_ASYNC(n) asm volatile("s_wait_asynccnt %0" ::"i"(n) : "memory")
#else
#define WAIT_ASYNC(n) ((void)0)
#endif

// Each lane copies 16B global -> its own LDS slot. With the async engine the
// transfer lands in LDS without a VGPR destination; we sync with ASYNCcnt
// only (each lane consumes only its own slot, so no block barrier needed).
__device__ __forceinline__ void tile_load16(const float* gp, float* lp) {
#if defined(HAVE_ASYNC_B128)
  __builtin_amdgcn_global_load_async_to_lds_b128(
      (__attribute__((address_space(1))) v4i*)gp,
      (__attribute__((address_space(3))) v4i*)lp, 0, 0);
#else
  *(v4f*)lp = *(const v4f*)gp;   // fallback: through-VGPR copy
#endif
}

// ---- deterministic per-element Gumbel via integer hash ------------------
__device__ __forceinline__ uint32_t hash_u32(uint32_t x) {
  x ^= x >> 17; x *= 0xed5ad4bbu;
  x ^= x >> 11; x *= 0xac4c1b51u;
  x ^= x >> 15; x *= 0x31848babu;
  x ^= x >> 14;
  return x;
}

// key = log|x| + Gumbel(hash(gid)).  The per-row -log(S) shift is constant,
// so top-k on this key equals top-k on log(prob)+gumbel.
__device__ __forceinline__ float key_of(float xval, uint32_t gid) {
  uint32_t h = hash_u32(gid ^ 0x9E3779B9u);
  float u01 = (float)((h >> 8) + 1u) * (1.0f / 16777218.0f);  // (0,1)
  float g = -__logf(-__logf(u01));
  return __logf(fabsf(xval)) + g;                  // |x|=0 -> -inf -> never picked
}

// monotone float -> uint32 order map
__device__ __forceinline__ uint32_t ord_of(float key) {
  uint32_t fu = __float_as_uint(key);
  return fu ^ (((uint32_t)((int32_t)fu >> 31)) | 0x80000000u);
}

// ---- kernel 0: zero the global histograms -------------------------------
__global__ void k_zero_hist(uint32_t* __restrict__ hist) {
  hist[(size_t)blockIdx.x * blockDim.x + threadIdx.x] = 0u;
}

// ---- kernel 1: fused row-sum(|x|) + key histogram (async-LDS pipeline) --
__global__ void __launch_bounds__(THREADS)
k_sum_hist(const float* __restrict__ x, uint32_t* __restrict__ hist,
           float* __restrict__ partial) {
  __shared__ __align__(16) float tile[2][ELEMS_PER_ITER];  // 8 KB
  __shared__ uint32_t lhist[NBINS];                        // 16 KB
  __shared__ float red[THREADS];                           // 1 KB

  const int tid = threadIdx.x;
  const int b   = blockIdx.x / CHUNKS1;
  const int c   = blockIdx.x % CHUNKS1;
  const uint32_t base = (uint32_t)b * N_ROW + (uint32_t)c * CHUNK1;

  for (int i = tid; i < NBINS; i += THREADS) lhist[i] = 0u;
  __syncthreads();

  const float* gp0 = x + base;
  tile_load16(gp0 + tid * VEC, &tile[0][tid * VEC]);

  float sum = 0.0f;
  for (int it = 0; it < ITERS1; ++it) {
    const int buf = it & 1;
    if (it + 1 < ITERS1) {
      tile_load16(gp0 + (it + 1) * ELEMS_PER_ITER + tid * VEC,
                  &tile[buf ^ 1][tid * VEC]);
      if (it + 2 < ITERS1)
        __builtin_prefetch(gp0 + (it + 2) * ELEMS_PER_ITER + tid * VEC, 0, 0);
      WAIT_ASYNC(1);   // oldest (current tile) has landed
    } else {
      WAIT_ASYNC(0);
    }
    v4f v = *(const v4f*)&tile[buf][tid * VEC];
    const uint32_t gidBase = base + (uint32_t)it * ELEMS_PER_ITER + tid * VEC;
#pragma unroll
    for (int j = 0; j < VEC; ++j) {
      float xv = v[j];
      sum += fabsf(xv);
      uint32_t u = ord_of(key_of(xv, gidBase + j));
      atomicAdd(&lhist[u >> BIN_SHIFT], 1u);     // integer: order-independent
    }
  }

  // fixed-order block tree reduction -> bit-deterministic partial sums
  red[tid] = sum;
  __syncthreads();
  for (int s = THREADS / 2; s > 0; s >>= 1) {
    if (tid < s) red[tid] += red[tid + s];
    __syncthreads();
  }
  if (tid == 0) partial[b * CHUNKS1 + c] = red[0];

  uint32_t* gh = hist + (size_t)b * NBINS;
  for (int i = tid; i < NBINS; i += THREADS) {
    uint32_t v = lhist[i];
    if (v) atomicAdd(&gh[i], v);
  }
}

// ---- kernel 2: per-row threshold (suffix scan) + 1/S --------------------
__global__ void __launch_bounds__(THREADS)
k_threshold(const uint32_t* __restrict__ hist, const float* __restrict__ partial,
            float* __restrict__ invS, uint32_t* __restrict__ thrU) {
  __shared__ uint32_t segSum[THREADS];
  const int b = blockIdx.x;
  const int tid = threadIdx.x;
  const uint32_t* gh = hist + (size_t)b * NBINS;
  const int SEG = NBINS / THREADS;  // 16 bins per thread

  uint32_t s = 0;
  for (int i = 0; i < SEG; ++i) s += gh[tid * SEG + i];
  segSum[tid] = s;
  __syncthreads();

  if (tid == 0) {
    float S = 0.0f;                               // fixed-order chunk sum
    for (int c = 0; c < CHUNKS1; ++c) S += partial[b * CHUNKS1 + c];
    invS[b] = 1.0f / S;

    uint32_t acc = 0;
    uint32_t thr = 0u;
    bool found = false;
    for (int seg = THREADS - 1; seg >= 0 && !found; --seg) {
      if (acc + segSum[seg] >= (uint32_t)K_SEL) {
        for (int i = seg * SEG + SEG - 1; i >= seg * SEG; --i) {
          acc += gh[i];
          if (acc >= (uint32_t)K_SEL) {
            thr = (uint32_t)i << BIN_SHIFT;
            found = true;
            break;
          }
        }
      } else {
        acc += segSum[seg];
      }
    }
    thrU[b] = thr;   // select iff ordered_key >= thr  (count >= K)
  }
}

// ---- kernel 3: select + scale + streaming NT store ----------------------
__global__ void __launch_bounds__(THREADS)
k_select_scale(const float* __restrict__ x, const float* __restrict__ invS,
               const uint32_t* __restrict__ thrU, float* __restrict__ out) {
  __shared__ __align__(16) float tile[2][ELEMS_PER_ITER];
  const int tid = threadIdx.x;
  const int b = blockIdx.x / CHUNKS3;
  const int c = blockIdx.x % CHUNKS3;
  const uint32_t base = (uint32_t)b * N_ROW + (uint32_t)c * CHUNK3;
  const float is = invS[b];
  const uint32_t thr = thrU[b];

  const float* gp0 = x + base;
  tile_load16(gp0 + tid * VEC, &tile[0][tid * VEC]);

  for (int it = 0; it < ITERS3; ++it) {
    const int buf = it & 1;
    if (it + 1 < ITERS3) {
      tile_load16(gp0 + (it + 1) * ELEMS_PER_ITER + tid * VEC,
                  &tile[buf ^ 1][tid * VEC]);
      if (it + 2 < ITERS3)
        __builtin_prefetch(gp0 + (it + 2) * ELEMS_PER_ITER + tid * VEC, 0, 0);
      WAIT_ASYNC(1);
    } else {
      WAIT_ASYNC(0);
    }
    v4f v = *(const v4f*)&tile[buf][tid * VEC];
    const uint32_t gidBase = base + (uint32_t)it * ELEMS_PER_ITER + tid * VEC;
    v4f o;
#pragma unroll
    for (int j = 0; j < VEC; ++j) {
      float xv = v[j];
      uint32_t u = ord_of(key_of(xv, gidBase + j));
      float res = 0.0f;
      if (u >= thr) {
        float p = fabsf(xv) * is;                          // prob
        float t = __expf((float)K_SEL * log1pf(-p));       // (1-p)^k
        res = xv / (1.0f - t + 1e-8f);                     // scale * x
      }
      o[j] = res;
    }
    // NT store: keep x resident in the 192MB L2 for this second pass
    __builtin_nontemporal_store(o, (v4f*)(out + gidBase));
  }
}

// ---- launcher -----------------------------------------------------------
extern "C" void kernel_launch(void* const* d_in, const int* in_sizes, int n_in,
                              void* d_out, int out_size, void* d_ws, size_t ws_size,
                              hipStream_t stream) {
  const float* x = (const float*)d_in[0];
  float* out = (float*)d_out;

  // workspace layout: hist[B][NBINS] u32 | partial[B][CHUNKS1] f32 | invS[B] | thrU[B]
  char* ws = (char*)d_ws;
  size_t off = 0;
  uint32_t* hist = (uint32_t*)(ws + off);  off += (size_t)B_DIM * NBINS * sizeof(uint32_t);
  float* partial = (float*)(ws + off);     off += (size_t)B_DIM * CHUNKS1 * sizeof(float);
  float* invS    = (float*)(ws + off);     off += (size_t)B_DIM * sizeof(float);
  uint32_t* thrU = (uint32_t*)(ws + off);

  k_zero_hist<<<(B_DIM * NBINS) / THREADS, THREADS, 0, stream>>>(hist);
  k_sum_hist<<<B_DIM * CHUNKS1, THREADS, 0, stream>>>(x, hist, partial);
  k_threshold<<<B_DIM, THREADS, 0, stream>>>(hist, partial, invS, thrU);
  k_select_scale<<<B_DIM * CHUNKS3, THREADS, 0, stream>>>(x, invS, thrU, out);
}